// NN_loss_51127290692352
// MI455X (gfx1250) — compile-verified
//
#include <hip/hip_runtime.h>
#include <math.h>

// ---------------------------------------------------------------------------
// NN_loss for pred/target (B=8,T=16,N=65536,3) fp32.
// out[s,c] = sqrt( min_d ( ra[c] - 2*mul[c][d] + rb[d] ) )  per slice s (128).
// Bandwidth-bound: 201 MB streamed once -> ~8.6 us floor @ 23.3 TB/s.
// Main loop: coalesced loads + 15 register FMAs/sample.
// WG reduction: ones(16x4) x LDSrows(4x16) via V_WMMA_F32_16X16X4_F32 chain.
// ---------------------------------------------------------------------------

typedef __attribute__((ext_vector_type(2))) float v2f;
typedef __attribute__((ext_vector_type(8))) float v8f;

#define BT_SLICES 128
#define NPTS      65536
#define SPLITS    4
#define THREADS   256
#define SAMPLES_PER_WG (NPTS / SPLITS)        // 16384
#define ITERS          (SAMPLES_PER_WG / THREADS) // 64

__global__ __launch_bounds__(THREADS)
void nnloss_gram_kernel(const float* __restrict__ pred,
                        const float* __restrict__ target,
                        float* __restrict__ partial)
{
    __shared__ float lds[THREADS * 16];

    const int tid   = threadIdx.x;
    const int wg    = blockIdx.x;
    const int split = wg & (SPLITS - 1);
    const int s     = wg >> 2;                 // log2(SPLITS)

    const float* __restrict__ p = pred   + (size_t)s * NPTS * 3;
    const float* __restrict__ t = target + (size_t)s * NPTS * 3;

    float ra0 = 0.f, ra1 = 0.f, ra2 = 0.f;
    float rb0 = 0.f, rb1 = 0.f, rb2 = 0.f;
    float m00 = 0.f, m01 = 0.f, m02 = 0.f;
    float m10 = 0.f, m11 = 0.f, m12 = 0.f;
    float m20 = 0.f, m21 = 0.f, m22 = 0.f;

    const int base = split * SAMPLES_PER_WG + tid;
#pragma unroll 4
    for (int i = 0; i < ITERS; ++i) {
        const size_t off = (size_t)(base + i * THREADS) * 3;
        // stream-ahead prefetch (speculative; OOB silently dropped)
        __builtin_prefetch(&p[off + (size_t)THREADS * 3 * 8], 0, 0);
        __builtin_prefetch(&t[off + (size_t)THREADS * 3 * 8], 0, 0);
        const float px = p[off + 0], py = p[off + 1], pz = p[off + 2];
        const float tx = t[off + 0], ty = t[off + 1], tz = t[off + 2];
        ra0 = fmaf(px, px, ra0); ra1 = fmaf(py, py, ra1); ra2 = fmaf(pz, pz, ra2);
        rb0 = fmaf(tx, tx, rb0); rb1 = fmaf(ty, ty, rb1); rb2 = fmaf(tz, tz, rb2);
        m00 = fmaf(px, tx, m00); m01 = fmaf(px, ty, m01); m02 = fmaf(px, tz, m02);
        m10 = fmaf(py, tx, m10); m11 = fmaf(py, ty, m11); m12 = fmaf(py, tz, m12);
        m20 = fmaf(pz, tx, m20); m21 = fmaf(pz, ty, m21); m22 = fmaf(pz, tz, m22);
    }

    // Per-thread row of the 256x16 accumulator matrix.
    float* row = &lds[tid * 16];
    row[0]  = ra0; row[1]  = ra1; row[2]  = ra2;
    row[3]  = rb0; row[4]  = rb1; row[5]  = rb2;
    row[6]  = m00; row[7]  = m01; row[8]  = m02;
    row[9]  = m10; row[10] = m11; row[11] = m12;
    row[12] = m20; row[13] = m21; row[14] = m22;
    row[15] = 0.f;

    __syncthreads();

    // Wave 0 reduces columns of the 256x16 matrix as ones(16x4) x B(4x16)
    // accumulated over 64 WMMAs. With A == ones, D[m][col] = sum_k B[k][col]
    // for every row m, so the result is invariant to B's slot->K mapping;
    // columns are striped lane&15 per the ISA B/C/D layout.
    if (tid < 32) {
        const int col = tid & 15;
        const int hi  = tid >> 4;   // lane half selects which rows this lane feeds
        v8f c = {0.f, 0.f, 0.f, 0.f, 0.f, 0.f, 0.f, 0.f};
        v2f a;
        a[0] = 1.0f;
        a[1] = 1.0f;
        for (int r = 0; r < THREADS; r += 4) {
            v2f b;
            b[0] = lds[(r + 2 * hi + 0) * 16 + col];
            b[1] = lds[(r + 2 * hi + 1) * 16 + col];
            // 8 args: (neg_a, A, neg_b, B, c_mod, C, reuse_a, reuse_b)
            c = __builtin_amdgcn_wmma_f32_16x16x4_f32(false, a, false, b,
                                                      (short)0, c, false, false);
        }
        if (tid < 16) {
            partial[(size_t)wg * 16 + tid] = c[0];  // lane n holds column-sum n
        }
    }
}

__global__ __launch_bounds__(128)
void nnloss_finish_kernel(const float* __restrict__ partial,
                          float* __restrict__ out)
{
    const int s = blockIdx.x * blockDim.x + threadIdx.x;
    if (s >= BT_SLICES) return;

    float g[15];
#pragma unroll
    for (int i = 0; i < 15; ++i) {
        float v = 0.f;
#pragma unroll
        for (int k = 0; k < SPLITS; ++k)
            v += partial[(size_t)(s * SPLITS + k) * 16 + i];
        g[i] = v;
    }

    // g[0..2]=ra, g[3..5]=rb, g[6+3c+d]=mul[c][d]
#pragma unroll
    for (int cc = 0; cc < 3; ++cc) {
        const float rac = g[cc];
        const float d0 = rac - 2.f * g[6 + 3 * cc + 0] + g[3];
        const float d1 = rac - 2.f * g[6 + 3 * cc + 1] + g[4];
        const float d2 = rac - 2.f * g[6 + 3 * cc + 2] + g[5];
        const float m  = fminf(d0, fminf(d1, d2));
        out[s * 3 + cc] = sqrtf(fmaxf(m, 0.f));  // sqrt(min) == norm at argmin
    }
}

extern "C" void kernel_launch(void* const* d_in, const int* in_sizes, int n_in,
                              void* d_out, int out_size, void* d_ws, size_t ws_size,
                              hipStream_t stream)
{
    (void)in_sizes; (void)n_in; (void)out_size; (void)ws_size;

    const float* pred   = (const float*)d_in[0];
    const float* target = (const float*)d_in[1];
    float*       out    = (float*)d_out;
    float*       partial = (float*)d_ws;   // BT_SLICES*SPLITS*16 floats = 32 KB

    nnloss_gram_kernel<<<BT_SLICES * SPLITS, THREADS, 0, stream>>>(pred, target, partial);
    nnloss_finish_kernel<<<1, 128, 0, stream>>>(partial, out);
}